// KNN_55473797595560
// MI455X (gfx1250) — compile-verified
//
#include <hip/hip_runtime.h>
#include <hip/hip_bf16.h>

typedef float v2f __attribute__((ext_vector_type(2)));
typedef float v8f __attribute__((ext_vector_type(8)));

#define N_TRAIN 32768
#define N_TEST  4096
#define DIM     64
#define KNN_K   32
#define NCLASS  1000
#define TILE_M  16
#define WAVES   8
#define NTILES  (N_TRAIN / 16)     // 2048 column tiles
#define ITERS   (NTILES / WAVES)   // 256 tiles per wave

__global__ __launch_bounds__(256)
void knn_wmma_kernel(const float* __restrict__ xtrain,
                     const int*   __restrict__ ytrain,
                     const float* __restrict__ xtest,
                     const float* __restrict__ wtrain,
                     float*       __restrict__ out)
{
    // per-(wave,row) top-K candidate lists
    __shared__ float listVal[WAVES * 16 * KNN_K];   // 16 KB
    __shared__ int   listIdx[WAVES * 16 * KNN_K];   // 16 KB
    __shared__ float stage[WAVES * 32];             // 1 KB score staging
    __shared__ float votes[16 * NCLASS];            // 62.5 KB

    const int tid  = threadIdx.x;
    const int lane = tid & 31;
    const int wv   = tid >> 5;
    const int hi   = lane >> 4;     // 0 or 1 (half-wave)
    const int lo   = lane & 15;

    // zero the vote table (block-wide)
    for (int i = tid; i < 16 * NCLASS; i += 256) votes[i] = 0.0f;
    // init this wave's candidate lists (wave-private, no barrier needed yet)
    for (int r = 0; r < 16; ++r) {
        listVal[(wv * 16 + r) * KNN_K + lane] = __builtin_inff();
        listIdx[(wv * 16 + r) * KNN_K + lane] = 0;
    }

    // ---- Hoist A fragment (16 test rows, scaled by -2) into registers ----
    // fp32 16x4 A layout: lane m=lane&15 holds row m; VGPR0/1 = K {0,1} for
    // lanes 0-15 and K {2,3} for lanes 16-31 (per K-step of 4).
    const int rowBase = blockIdx.x * TILE_M;
    float a0[16], a1[16];
    {
        const float* ap = xtest + (size_t)(rowBase + lo) * DIM + 2 * hi;
        #pragma unroll
        for (int t = 0; t < 16; ++t) {
            float2 av = *reinterpret_cast<const float2*>(ap + 4 * t);
            a0[t] = -2.0f * av.x;
            a1[t] = -2.0f * av.y;
        }
    }

    // running per-row worst (rows v + 8*hi), kept coherent via shfl broadcast
    float w8[8];
    #pragma unroll
    for (int v = 0; v < 8; ++v) w8[v] = __builtin_inff();

    // ---- Main loop: each wave sweeps 256 column tiles of 16 train rows ----
    for (int it = 0; it < ITERS; ++it) {
        const int nt      = wv + WAVES * it;
        const int colBase = nt * 16;
        const int col     = colBase + lo;
        const float* bp   = xtrain + (size_t)col * DIM + 2 * hi;

        v8f acc = {};       // C starts at 0 -> acc = -2 * (a . b)
        float p = 0.0f;     // partial |train row|^2 (this lane's half)
        #pragma unroll
        for (int t = 0; t < 16; ++t) {
            float2 bvv = *reinterpret_cast<const float2*>(bp + 4 * t);
            p = fmaf(bvv.x, bvv.x, p);
            p = fmaf(bvv.y, bvv.y, p);
            v2f A; A.x = a0[t];  A.y = a1[t];
            v2f B; B.x = bvv.x;  B.y = bvv.y;
            acc = __builtin_amdgcn_wmma_f32_16x16x4_f32(
                false, A, false, B, (short)0, acc, false, false);
        }
        // full train norm: this half + partner half (same column, lane^16)
        const float b2 = p + __shfl_xor(p, 16, 32);

        // ---- top-K update: score = |b|^2 - 2 a.b  (rank == distance rank)
        #pragma unroll
        for (int v = 0; v < 8; ++v) {
            const float sv = acc[v] + b2;       // row = v + 8*hi, column = col
            float wst = w8[v];
            unsigned long long bal = __ballot(sv < wst);
            if (bal != 0ull) {                  // uniform across the wave
                stage[wv * 32 + lane] = sv;
                __builtin_amdgcn_wave_barrier();
                asm volatile("s_wait_dscnt 0" ::: "memory");
                if (lane == 0 || lane == 16) {  // one inserter per half/row
                    unsigned mm = (lane == 0)
                        ? (unsigned)(bal & 0xFFFFu)
                        : (unsigned)((bal >> 16) & 0xFFFFu);
                    const int row    = v + ((lane == 16) ? 8 : 0);
                    const int srcOfs = (lane == 16) ? 16 : 0;
                    float* LV = &listVal[(wv * 16 + row) * KNN_K];
                    int*   LI = &listIdx[(wv * 16 + row) * KNN_K];
                    while (mm) {
                        const int b = __ffs(mm) - 1; mm &= mm - 1;
                        const float val = stage[wv * 32 + srcOfs + b];
                        if (val < wst) {
                            int wsl = 0; float wvv = LV[0];
                            for (int i = 1; i < KNN_K; ++i)
                                if (LV[i] > wvv) { wvv = LV[i]; wsl = i; }
                            LV[wsl] = val;
                            LI[wsl] = colBase + b;
                            wvv = LV[0];
                            for (int i = 1; i < KNN_K; ++i)
                                if (LV[i] > wvv) wvv = LV[i];
                            wst = wvv;
                        }
                    }
                }
                __builtin_amdgcn_wave_barrier();
                // broadcast updated threshold from lane 0 / lane 16
                w8[v] = __shfl(wst, lane & 16, 32);
            }
        }
    }

    __syncthreads();

    // ---- Merge 8 wave-lists (256 candidates) -> top-32, vote ----
    // wave wv handles rows 2*wv and 2*wv+1
    for (int rr = 0; rr < 2; ++rr) {
        const int r = 2 * wv + rr;
        float cv[8]; int ci[8];
        #pragma unroll
        for (int j = 0; j < 8; ++j) {
            cv[j] = listVal[(j * 16 + r) * KNN_K + lane];
            ci[j] = listIdx[(j * 16 + r) * KNN_K + lane];
        }
        for (int k = 0; k < KNN_K; ++k) {
            float bvv = cv[0]; int bi = ci[0]; int ls = 0;
            #pragma unroll
            for (int j = 1; j < 8; ++j)
                if (cv[j] < bvv) { bvv = cv[j]; bi = ci[j]; ls = j; }
            int bl = lane;
            #pragma unroll
            for (int off = 16; off > 0; off >>= 1) {
                const float ov = __shfl_xor(bvv, off, 32);
                const int   oi = __shfl_xor(bi,  off, 32);
                const int   ol = __shfl_xor(bl,  off, 32);
                if (ov < bvv || (ov == bvv && ol < bl)) { bvv = ov; bi = oi; bl = ol; }
            }
            if (lane == bl) cv[ls] = __builtin_inff();   // remove winner
            if (lane == 0) {                              // weighted vote
                const int lbl = ytrain[bi];
                votes[r * NCLASS + lbl] += wtrain[bi];
            }
        }
    }

    __syncthreads();

    // ---- argmax over classes (first-occurrence ties, like jnp.argmax) ----
    for (int rr = 0; rr < 2; ++rr) {
        const int r = 2 * wv + rr;
        float bvv = votes[r * NCLASS + lane]; int bc = lane;
        for (int c = lane + 32; c < NCLASS; c += 32) {
            const float vv = votes[r * NCLASS + c];
            if (vv > bvv) { bvv = vv; bc = c; }
        }
        #pragma unroll
        for (int off = 16; off > 0; off >>= 1) {
            const float ov = __shfl_xor(bvv, off, 32);
            const int   oc = __shfl_xor(bc,  off, 32);
            if (ov > bvv || (ov == bvv && oc < bc)) { bvv = ov; bc = oc; }
        }
        if (lane == 0) out[rowBase + r] = (float)bc;
    }
}

extern "C" void kernel_launch(void* const* d_in, const int* in_sizes, int n_in,
                              void* d_out, int out_size, void* d_ws, size_t ws_size,
                              hipStream_t stream) {
    const float* xtrain = (const float*)d_in[0];   // [32768, 64] f32
    const int*   ytrain = (const int*)d_in[1];     // [32768] int
    const float* xtest  = (const float*)d_in[2];   // [4096, 64] f32
    const float* wtrain = (const float*)d_in[3];   // [32768] f32
    float* out = (float*)d_out;                    // [4096] labels as float

    knn_wmma_kernel<<<N_TEST / TILE_M, 256, 0, stream>>>(
        xtrain, ytrain, xtest, wtrain, out);
}